// PPIEncdoer_49435073577269
// MI455X (gfx1250) — compile-verified
//
#include <hip/hip_runtime.h>
#include <hip/hip_bf16.h>

#define NNODES 50000
#define IN_C   512
#define OUT_C  200   // per conv
#define TOT_C  400   // both convs concatenated

typedef __attribute__((ext_vector_type(2))) float v2f;
typedef __attribute__((ext_vector_type(8))) float v8f;

// ---------------- degree / normalization ----------------

__global__ __launch_bounds__(256) void zero_f32_kernel(float* __restrict__ p, int n) {
    int i = blockIdx.x * blockDim.x + threadIdx.x;
    if (i < n) p[i] = 0.0f;
}

__global__ __launch_bounds__(256) void deg_count_kernel(const long long* __restrict__ dst,
                                                        float* __restrict__ deg, int E) {
    int e = blockIdx.x * blockDim.x + threadIdx.x;
    if (e < E) atomicAdd(&deg[(int)dst[e]], 1.0f);
}

__global__ __launch_bounds__(256) void make_dinv_kernel(float* __restrict__ deg, int n) {
    int i = blockIdx.x * blockDim.x + threadIdx.x;
    if (i < n) deg[i] = rsqrtf(deg[i] + 1.0f);   // +1 = self loop; always > 0
}

// ---------------- fused dual GEMM: H[N,400] = x @ [W_mu | W_ls] ----------------
// One wave per 16x16 output tile. M tiles: 50000/16 = 3125 (exact).
// N tiles: 400/16 = 25 (exact). K = 512 -> 128 x V_WMMA_F32_16X16X4_F32.

__global__ __launch_bounds__(32) void gemm_wmma_f32_kernel(
    const float* __restrict__ x,       // [N, 512]
    const float* __restrict__ Wmu,     // [512, 200] row-major (k, n)
    const float* __restrict__ Wls,     // [512, 200]
    float* __restrict__ H)             // [N, 400]
{
    const int lane  = threadIdx.x;        // 0..31
    const int half  = lane >> 4;          // 0: lanes 0-15, 1: lanes 16-31
    const int l16   = lane & 15;
    const int mbase = blockIdx.x * 16;
    const int nbase = blockIdx.y * 16;
    const int col   = nbase + l16;        // 0..399, exact

    // Column pointer into the logical concatenated weight [512,400]
    const float* __restrict__ Wp = (col < OUT_C) ? (Wmu + col) : (Wls + (col - OUT_C));
    // A fragment base: row = mbase + l16; lanes>=16 carry K+2,K+3
    const float* __restrict__ Arow = x + (size_t)(mbase + l16) * IN_C + 2 * half;

    v8f acc = {};
    #pragma unroll 8
    for (int k0 = 0; k0 < IN_C; k0 += 4) {
        // A 16x4 f32: VGPR0=K{0|2}, VGPR1=K{1|3} -> contiguous float2 per lane
        v2f a = *(const v2f*)(Arow + k0);
        // B 4x16 f32: VGPR0=row K{0|2} striped over lanes, VGPR1=row K{1|3}
        const int kb = k0 + 2 * half;
        v2f b;
        b.x = Wp[(size_t)kb * OUT_C];
        b.y = Wp[(size_t)(kb + 1) * OUT_C];
        acc = __builtin_amdgcn_wmma_f32_16x16x4_f32(
            /*neg_a=*/false, a, /*neg_b=*/false, b,
            /*c_mod=*/(short)0, acc, /*reuse_a=*/false, /*reuse_b=*/false);
    }

    // D 16x16 f32: VGPR g -> M = g + 8*half, N = l16
    #pragma unroll
    for (int g = 0; g < 8; ++g) {
        const int row = mbase + g + 8 * half;
        H[(size_t)row * TOT_C + col] = acc[g];
    }
}

// ---------------- output init: bias + self-loop contribution ----------------
// out layout: mu flat [N,200] then logstd flat [N,200].

__global__ __launch_bounds__(256) void init_out_kernel(
    const float* __restrict__ H, const float* __restrict__ dinv,
    const float* __restrict__ bmu, const float* __restrict__ bls,
    float* __restrict__ out)
{
    size_t i = (size_t)blockIdx.x * blockDim.x + threadIdx.x;  // over N*400
    if (i >= (size_t)NNODES * TOT_C) return;
    const int node = (int)(i / TOT_C);
    const int c    = (int)(i % TOT_C);
    const float di = dinv[node];
    const float selfw = di * di;                 // self-loop norm
    const float bias = (c < OUT_C) ? bmu[c] : bls[c - OUT_C];
    const float v = H[i] * selfw + bias;
    const size_t o = (c < OUT_C)
        ? ((size_t)node * OUT_C + c)
        : ((size_t)NNODES * OUT_C + (size_t)node * OUT_C + (c - OUT_C));
    out[o] = v;
}

// ---------------- edge scatter: out[dst] += H[src] * dinv[src]*dinv[dst] ----------------
// One wave per edge; lanes stride the 400 channels (coalesced 128B gathers,
// float atomics resolved at the L2 atomic units).

__global__ __launch_bounds__(256) void edge_scatter_kernel(
    const long long* __restrict__ src, const long long* __restrict__ dst,
    const float* __restrict__ H, const float* __restrict__ dinv,
    float* __restrict__ out, int E)
{
    const int wave = (int)((blockIdx.x * (size_t)blockDim.x + threadIdx.x) >> 5);
    const int lane = threadIdx.x & 31;
    if (wave >= E) return;
    const int s = (int)src[wave];
    const int d = (int)dst[wave];
    const float w = dinv[s] * dinv[d];
    const float* __restrict__ hs = H + (size_t)s * TOT_C;
    float* __restrict__ out_mu = out + (size_t)d * OUT_C;
    float* __restrict__ out_ls = out + (size_t)NNODES * OUT_C + (size_t)d * OUT_C;
    #pragma unroll
    for (int c = lane; c < TOT_C; c += 32) {
        const float v = hs[c] * w;
        if (c < OUT_C) atomicAdd(out_mu + c, v);
        else           atomicAdd(out_ls + (c - OUT_C), v);
    }
}

// ---------------- launch ----------------

extern "C" void kernel_launch(void* const* d_in, const int* in_sizes, int n_in,
                              void* d_out, int out_size, void* d_ws, size_t ws_size,
                              hipStream_t stream) {
    const float*      x    = (const float*)d_in[0];      // [N,512]
    const float*      Wmu  = (const float*)d_in[1];      // [512,200]
    const float*      bmu  = (const float*)d_in[2];      // [200]
    const float*      Wls  = (const float*)d_in[3];      // [512,200]
    const float*      bls  = (const float*)d_in[4];      // [200]
    const long long*  ei   = (const long long*)d_in[5];  // [2,E] int64
    const int E = in_sizes[5] / 2;
    const long long* src = ei;
    const long long* dst = ei + E;

    float* out = (float*)d_out;                          // mu[N,200] ++ logstd[N,200]

    // workspace: dinv (N floats) then H (N*400 floats), 256B-aligned offset
    float* ws   = (float*)d_ws;
    float* dinv = ws;
    float* H    = ws + 51200;                            // 51200*4 bytes in

    // 1) degree with self-loops -> dinv
    zero_f32_kernel<<<(NNODES + 255) / 256, 256, 0, stream>>>(dinv, NNODES);
    deg_count_kernel<<<(E + 255) / 256, 256, 0, stream>>>(dst, dinv, E);
    make_dinv_kernel<<<(NNODES + 255) / 256, 256, 0, stream>>>(dinv, NNODES);

    // 2) fused dual GEMM via f32 WMMA
    dim3 ggrid(NNODES / 16, TOT_C / 16);                 // 3125 x 25, exact
    gemm_wmma_f32_kernel<<<ggrid, 32, 0, stream>>>(x, Wmu, Wls, H);

    // 3) out = bias + self-loop term
    const size_t tot = (size_t)NNODES * TOT_C;
    init_out_kernel<<<(unsigned)((tot + 255) / 256), 256, 0, stream>>>(H, dinv, bmu, bls, out);

    // 4) edge scatter-add
    const int waves_per_block = 256 / 32;
    edge_scatter_kernel<<<(E + waves_per_block - 1) / waves_per_block, 256, 0, stream>>>(
        src, dst, H, dinv, out, E);
}